// SAGPool_15917148799734
// MI455X (gfx1250) — compile-verified
//
#include <hip/hip_runtime.h>
#include <math.h>

// Problem constants (from the reference)
#define N_NODES 50000
#define N_EDGES 600000
#define CH      128
#define K_OUT   25000            // int(50000 * 0.5)
#define N_TILES (N_NODES / 16)   // 3125, exact
#define RTILE   1024             // rank-kernel LDS tile (floats); 50000%4==0, RTILE%4==0

typedef __attribute__((ext_vector_type(2))) float v2f;
typedef __attribute__((ext_vector_type(8))) float v8f;
typedef __attribute__((ext_vector_type(4))) int   v4i;
typedef __attribute__((address_space(1))) v4i     gv4i;  // global int4
typedef __attribute__((address_space(3))) v4i     lv4i;  // LDS int4

// ---------------------------------------------------------------------------
// gfx1250 async copy global->LDS (ASYNCcnt-tracked).
// Builtin signature (from hipcc diagnostic): (v4i AS1*, v4i AS3*, Iimm, Icpol)
// ---------------------------------------------------------------------------
__device__ __forceinline__ void async_copy_b128(const float* g, float* l) {
#if defined(__has_builtin) && __has_builtin(__builtin_amdgcn_global_load_async_to_lds_b128)
  v4i* gp_generic = (v4i*)const_cast<float*>(g);
  v4i* lp_generic = (v4i*)l;
  __builtin_amdgcn_global_load_async_to_lds_b128(
      (gv4i*)gp_generic, (lv4i*)lp_generic, /*imm offset=*/0, /*cpol=*/0);
#else
  const unsigned lds_off =
      (unsigned)(size_t)(__attribute__((address_space(3))) char*)(void*)l;
  asm volatile("global_load_async_to_lds_b128 %0, %1, off"
               :: "v"(lds_off), "v"(g) : "memory");
#endif
}

__device__ __forceinline__ void wait_async0() {
#if defined(__has_builtin) && __has_builtin(__builtin_amdgcn_s_wait_asynccnt)
  __builtin_amdgcn_s_wait_asynccnt(0);
#else
  asm volatile("s_wait_asynccnt 0" ::: "memory");
#endif
}

// ---------------------------------------------------------------------------
// Stage 1: scores[i] = sigmoid(dot(x[i,:], W[0,:])) via V_WMMA_F32_16X16X4_F32
// One wave handles 16 nodes. A = 16x4 tile of x, B = 4x16 with every column
// equal to the W chunk, so D[m][n] == dot(x[m],W) for every n.
// A (32-bit 16x4): lanes 0-15 -> M=0..15 {K=0,K=1}; lanes 16-31 -> {K=2,K=3}.
// B (4x16): VGPR0 lanes0-15 = row K=0, lanes16-31 = row K=2; VGPR1 = K=1/K=3.
// With ksel = (lane>=16)*2, both A and B per-lane loads are float2 at [k0+ksel].
// C/D: VGPR j, lanes 0-15 -> D[M=j][N=lane], lanes 16-31 -> D[M=j+8][N=lane-16].
// So lane 0 holds scores of nodes base+0..7, lane 16 holds base+8..15.
// ---------------------------------------------------------------------------
__global__ __launch_bounds__(256) void scores_wmma_kernel(
    const float* __restrict__ x, const float* __restrict__ W,
    float* __restrict__ scores) {
  const int wave = threadIdx.x >> 5;
  const int lane = threadIdx.x & 31;
  const int tile = blockIdx.x * 8 + wave;
  if (tile >= N_TILES) return;  // wave-uniform branch: EXEC stays all-1s

  const int row  = tile * 16 + (lane & 15);
  const int ksel = (lane >> 4) << 1;  // 0 for lanes 0-15, 2 for lanes 16-31
  const float* xr = x + (size_t)row * CH;

  v8f c = {};
  #pragma unroll 4
  for (int k0 = 0; k0 < CH; k0 += 4) {
    const v2f a = *(const v2f*)(xr + k0 + ksel);   // A: x[row][k0+ksel .. +1]
    const v2f b = *(const v2f*)(W  + k0 + ksel);   // B: W broadcast to all 16 cols
    c = __builtin_amdgcn_wmma_f32_16x16x4_f32(
            /*neg_a=*/false, a, /*neg_b=*/false, b,
            /*c_mod=*/(short)0, c, /*reuse_a=*/false, /*reuse_b=*/false);
  }

  // Divergent tail is fine: all WMMAs already executed with full EXEC.
  if ((lane & 15) == 0) {
    const int mbase = tile * 16 + ((lane >> 4) << 3);  // lane0 -> +0, lane16 -> +8
    #pragma unroll
    for (int j = 0; j < 8; ++j) {
      const float z = c[j];
      scores[mbase + j] = 1.0f / (1.0f + __expf(-z));
    }
  }
}

// ---------------------------------------------------------------------------
// Stage 2: x_pool[dst] += x[src]  (edge softmax over a size-1 axis == 1.0)
// One wave per edge: 32 lanes x float4 = 128 channels, hardware f32 atomics.
// ---------------------------------------------------------------------------
__global__ __launch_bounds__(256) void scatter_kernel(
    const float* __restrict__ x, const int* __restrict__ ei,
    float* __restrict__ pool) {
  const int e = blockIdx.x * 8 + (threadIdx.x >> 5);
  if (e >= N_EDGES) return;
  const int lane = threadIdx.x & 31;
  const int src = ei[e];             // edge_index[0][e]
  const int dst = ei[N_EDGES + e];   // edge_index[1][e]

  const float4 v = ((const float4*)(x + (size_t)src * CH))[lane];
  float* p = pool + (size_t)dst * CH + lane * 4;
  unsafeAtomicAdd(p + 0, v.x);
  unsafeAtomicAdd(p + 1, v.y);
  unsafeAtomicAdd(p + 2, v.z);
  unsafeAtomicAdd(p + 3, v.w);
}

// ---------------------------------------------------------------------------
// Stage 3: exact descending top-k with jax.lax.top_k tie semantics.
// rank(i) = #{j : s_j > s_i} + #{j < i : s_j == s_i}; ranks are a permutation,
// so every slot in [0,K_OUT) is written exactly once.
// Double-buffered: tile t+1 is prefetched into LDS via async-to-LDS while
// comparing against tile t. Each wave waits its own ASYNCcnt, then the block
// barrier proves whole-tile visibility. Async over-read past the scores array
// lands in owned workspace and is never compared (lim4 bound).
// ---------------------------------------------------------------------------
__global__ __launch_bounds__(256) void rank_kernel(
    const float* __restrict__ scores, int* __restrict__ idx) {
  __shared__ __align__(16) float buf[2][RTILE];
  const int tid = threadIdx.x;
  const int i = blockIdx.x * 256 + tid;
  const float si = (i < N_NODES) ? scores[i] : 3.0f;  // sentinel; store is guarded

  // Prologue: prefetch tile 0 (256 threads x 16B = 4KB).
  async_copy_b128(scores + tid * 4, &buf[0][tid * 4]);

  const int ntiles = (N_NODES + RTILE - 1) / RTILE;  // 49
  int r = 0;
  for (int t = 0; t < ntiles; ++t) {
    wait_async0();      // own async loads done
    __syncthreads();    // everyone's async loads done -> tile t fully in LDS
    if (t + 1 < ntiles) // prefetch next tile; its buffer was released by the
                        // end-of-iteration barrier of t-1 (before this point)
      async_copy_b128(scores + (t + 1) * RTILE + tid * 4,
                      &buf[(t + 1) & 1][tid * 4]);

    const int base = t * RTILE;
    const int lim4 = min(RTILE, N_NODES - base) >> 2;  // valid length, /4 exact
    const float4* s4p = (const float4*)buf[t & 1];
    const int eqb = i - base;  // tie-break: global j < i  <=>  jj < eqb
    for (int q = 0; q < lim4; ++q) {
      const float4 s4 = s4p[q];
      const int jj = q << 2;
      r += (int)((s4.x > si) | ((s4.x == si) & (jj + 0 < eqb)));
      r += (int)((s4.y > si) | ((s4.y == si) & (jj + 1 < eqb)));
      r += (int)((s4.z > si) | ((s4.z == si) & (jj + 2 < eqb)));
      r += (int)((s4.w > si) | ((s4.w == si) & (jj + 3 < eqb)));
    }
    __syncthreads();    // all reads of buf[t&1] done before t+2 overwrites it
  }
  if (i < N_NODES && r < K_OUT) idx[r] = i;
}

// ---------------------------------------------------------------------------
// Stage 4: out[r,:] = pool[idx[r],:]   (one wave per output row)
// ---------------------------------------------------------------------------
__global__ __launch_bounds__(256) void gather_kernel(
    const float* __restrict__ pool, const int* __restrict__ idx,
    float* __restrict__ out) {
  const int r = blockIdx.x * 8 + (threadIdx.x >> 5);
  if (r >= K_OUT) return;
  const int lane = threadIdx.x & 31;
  const int node = idx[r];
  ((float4*)(out + (size_t)r * CH))[lane] =
      ((const float4*)(pool + (size_t)node * CH))[lane];
}

// ---------------------------------------------------------------------------
extern "C" void kernel_launch(void* const* d_in, const int* in_sizes, int n_in,
                              void* d_out, int out_size, void* d_ws, size_t ws_size,
                              hipStream_t stream) {
  const float* x  = (const float*)d_in[0];   // [50000,128] f32
  const float* W  = (const float*)d_in[1];   // [1,128]     f32
  const int*   ei = (const int*)d_in[2];     // [2,600000]  int
  float* out = (float*)d_out;                // [25000,128] f32

  // Workspace layout (256B aligned): pool | scores | idx  (~25.9 MB total)
  char* ws = (char*)d_ws;
  const size_t pool_bytes = (size_t)N_NODES * CH * sizeof(float);
  const size_t scores_off = (pool_bytes + 255) & ~(size_t)255;
  const size_t idx_off    = (scores_off + (size_t)N_NODES * sizeof(float) + 255) & ~(size_t)255;
  float* pool   = (float*)(ws);
  float* scores = (float*)(ws + scores_off);
  int*   idxbuf = (int*)(ws + idx_off);

  // Zero the accumulation buffer every call (ws is not re-poisoned between replays).
  (void)hipMemsetAsync(pool, 0, pool_bytes, stream);

  // 1) scores via WMMA (3125 tiles of 16 nodes, 8 waves/block)
  scores_wmma_kernel<<<(N_TILES + 7) / 8, 256, 0, stream>>>(x, W, scores);

  // 2) scatter-add messages (one wave per edge)
  scatter_kernel<<<(N_EDGES + 7) / 8, 256, 0, stream>>>(x, ei, pool);

  // 3) exact top-k ranking (async double-buffered LDS tiles)
  rank_kernel<<<(N_NODES + 255) / 256, 256, 0, stream>>>(scores, idxbuf);

  // 4) gather selected rows
  gather_kernel<<<(K_OUT + 7) / 8, 256, 0, stream>>>(pool, idxbuf, out);
}